// GRIPModel_15049565405565
// MI455X (gfx1250) — compile-verified
//
#include <hip/hip_runtime.h>

// ---------------------------------------------------------------------------
// GRIP model forward for MI455X (gfx1250): fp32 WMMA (16x16x4) everywhere,
// dense-in-LDS normalized adjacency, async global->LDS activation staging.
// ---------------------------------------------------------------------------

typedef __attribute__((ext_vector_type(2))) float v2f;
typedef __attribute__((ext_vector_type(8))) float v8f;

static constexpr int BB   = 32;
static constexpr int C0   = 4;
static constexpr int NN   = 256;
static constexpr int TT   = 16;
static constexpr int KNNK = 6;
static constexpr int SROW = 260;          // padded LDS row stride (floats)
static constexpr float ALPHA = 0.001f;

__device__ __forceinline__ v8f wmma_f32(v2f a, v2f b, v8f c) {
  // D = A(16x4, f32) * B(4x16, f32) + C(16x16, f32)
  return __builtin_amdgcn_wmma_f32_16x16x4_f32(
      /*neg_a=*/false, a, /*neg_b=*/false, b,
      /*c_mod=*/(short)0, c, /*reuse_a=*/false, /*reuse_b=*/false);
}

// Async global->LDS copy of one dword (ASYNCcnt-tracked, no VGPR round-trip).
__device__ __forceinline__ void async_copy_b32(unsigned lds_byte_off,
                                               const float* gaddr) {
  asm volatile("global_load_async_to_lds_b32 %0, %1, off"
               :: "v"(lds_byte_off),
                  "v"((unsigned long long)(uintptr_t)gaddr)
               : "memory");
}
__device__ __forceinline__ void wait_async() {
  asm volatile("s_wait_asynccnt 0" ::: "memory");
}

// ---------------------------------------------------------------------------
// Adjacency: per (b,t) KNN(K=6) over 256 nodes, degree, D^-1 and D^-1/2.
// knn[...] = neighbor index, or -1 when the row is masked out.
// ---------------------------------------------------------------------------
__global__ __launch_bounds__(256) void adj_kernel(
    const float* __restrict__ x, int* __restrict__ knn,
    float* __restrict__ dinv_g, float* __restrict__ dis_g) {
  __shared__ float sx[NN], sy[NN], sdeg[NN];
  const int gt = blockIdx.x;            // b*TT + t
  const int b  = gt / TT, t = gt % TT;
  const int i  = threadIdx.x;
  const float px = x[((b * C0 + 0) * NN + i) * TT + t];
  const float py = x[((b * C0 + 1) * NN + i) * TT + t];
  sx[i] = px; sy[i] = py; sdeg[i] = 0.0f;
  __syncthreads();

  float bd[KNNK]; int bj[KNNK];
#pragma unroll
  for (int k = 0; k < KNNK; ++k) { bd[k] = 3.4e38f; bj[k] = 0; }
  for (int j = 0; j < NN; ++j) {
    if (j == i) continue;
    const float dx = px - sx[j], dy = py - sy[j];
    float d = dx * dx + dy * dy;
    int   jj = j;
#pragma unroll
    for (int k = 0; k < KNNK; ++k) {
      if (d < bd[k]) {
        const float td = bd[k]; const int tj = bj[k];
        bd[k] = d; bj[k] = jj; d = td; jj = tj;
      }
    }
  }
  const bool mask = (px != 0.0f) && (py != 0.0f);
  if (mask) {
#pragma unroll
    for (int k = 0; k < KNNK; ++k) atomicAdd(&sdeg[bj[k]], 1.0f);
  }
  __syncthreads();
  const float dv = 1.0f / (sdeg[i] + ALPHA);
  dinv_g[gt * NN + i] = dv;
  dis_g [gt * NN + i] = sqrtf(dv);
#pragma unroll
  for (int k = 0; k < KNNK; ++k)
    knn[(gt * NN + i) * KNNK + k] = mask ? bj[k] : -1;
}

// ---------------------------------------------------------------------------
// Temporal 1x3 conv as implicit GEMM per (b, t_out):
//   out[co, n] = bias[co] + sum_k W[co, k] * X[k, n],  K = 3*Cin (k = ci*3+dt)
// 8 wave32 per block; each wave computes a 16(co) x 64(n) tile: one W fragment
// feeds 4 WMMAs. Zero-padding via clamped index + cndmask (no exec branches).
// ---------------------------------------------------------------------------
template <int CIN, int COUT, int STRIDE, int TIN, int TOUT>
__global__ __launch_bounds__(256) void conv_kernel(
    const float* __restrict__ x, const float* __restrict__ W,
    const float* __restrict__ bias, float* __restrict__ out) {
  constexpr int KT = 3 * CIN;
  const int bt   = blockIdx.x;
  const int b    = bt / TOUT, tout = bt % TOUT;
  const int lane = threadIdx.x & 31;
  const int wave = threadIdx.x >> 5;
  const int ln = lane & 15, lh = lane >> 4;
  const int tbase = tout * STRIDE - 1;
  constexpr int NGROUPS = (COUT / 16) * 4;     // co-tiles x 64-wide n-groups

  for (int g = wave; g < NGROUPS; g += 8) {
    const int co0 = (g >> 2) * 16;
    const int ng0 = (g & 3) * 64;
    v8f acc[4] = {};
    for (int k0 = 0; k0 < KT; k0 += 4) {
      const int ka = k0 + 2 * lh;
      const v2f a = *(const v2f*)&W[(co0 + ln) * KT + ka];   // A frag (b64)
      const int ci0 = ka / 3,      dt0 = ka - 3 * ci0;
      const int ci1 = (ka + 1) / 3, dt1 = (ka + 1) - 3 * ci1;
      const int t0 = tbase + dt0, t1 = tbase + dt1;
      const bool v0 = (unsigned)t0 < (unsigned)TIN;
      const bool v1 = (unsigned)t1 < (unsigned)TIN;
      const float* px0 = &x[(size_t)(b * CIN + ci0) * NN * TIN + (v0 ? t0 : 0)];
      const float* px1 = &x[(size_t)(b * CIN + ci1) * NN * TIN + (v1 ? t1 : 0)];
#pragma unroll
      for (int cc = 0; cc < 4; ++cc) {
        const int n = ng0 + cc * 16 + ln;
        const float r0 = px0[n * TIN];
        const float r1 = px1[n * TIN];
        v2f bf;
        bf.x = v0 ? r0 : 0.0f;
        bf.y = v1 ? r1 : 0.0f;
        acc[cc] = wmma_f32(a, bf, acc[cc]);
      }
    }
#pragma unroll
    for (int cc = 0; cc < 4; ++cc) {
#pragma unroll
      for (int r = 0; r < 8; ++r) {
        const int co = co0 + r + 8 * lh;       // D layout: M = r + 8*lh, N = ln
        const int n  = ng0 + cc * 16 + ln;
        out[((b * COUT + co) * NN + n) * TOUT + tout] = acc[cc][r] + bias[co];
      }
    }
  }
}

// ---------------------------------------------------------------------------
// Graph op per (b,t): out = (D^-1 + D^-1/2 A D^-1/2) * X  as 256x256xC GEMM.
//  - As: dense normalized adjacency, 128-row halves in LDS (row stride 260)
//  - Xs: activation panel staged transposed in LDS via async global->LDS, so
//    each B fragment is one ds_load_b64
//  - each wave computes a 16(i) x 64(c) tile: 1 A frag -> 4 WMMAs
// ---------------------------------------------------------------------------
template <int C, int TC>
__global__ __launch_bounds__(256) void graph_kernel(
    const float* __restrict__ xin, const int* __restrict__ knn,
    const float* __restrict__ dinv_g, const float* __restrict__ dis_g,
    float* __restrict__ out) {
  constexpr int CH  = (C > 128) ? 128 : C;     // channels resident per pass
  constexpr int NCH = C / CH;
  constexpr int CG  = CH / 64;                 // 64-wide c-groups per pass
  __shared__ float As[128 * SROW];
  __shared__ float Xs[CH * SROW];
  __shared__ float sdis[NN];
  const int bt = blockIdx.x;
  const int b = bt / TC, t = bt % TC;
  const int abase = (b * TT + t) * NN;         // adjacency arrays use T=16
  const int tid  = threadIdx.x;
  const int lane = tid & 31, wave = tid >> 5;
  const int ln = lane & 15, lh = lane >> 4;

  sdis[tid] = dis_g[abase + tid];

  for (int ch = 0; ch < NCH; ++ch) {
    // ---- stage Xs[c_local][j] = x[b, ch*CH+c_local, j, t] (async -> LDS)
    for (int idx = tid; idx < CH * NN; idx += 256) {
      const int cl = idx >> 8, j = idx & 255;
      async_copy_b32((unsigned)(uintptr_t)&Xs[cl * SROW + j],
                     &xin[((size_t)(b * C + ch * CH + cl) * NN + j) * TC + t]);
    }

    for (int h = 0; h < 2; ++h) {
      // ---- zero adjacency half (vectorized b128 stores)
      float4* As4 = (float4*)As;
      for (int idx = tid; idx < (128 * SROW) / 4; idx += 256)
        As4[idx] = make_float4(0.f, 0.f, 0.f, 0.f);
      __syncthreads();
      // ---- scatter normalized adjacency rows; fold D^-1 on the diagonal
      if (tid < 128) {
        const int i = h * 128 + tid;
        As[tid * SROW + i] = dinv_g[abase + i];
        const float di = sdis[i];
#pragma unroll
        for (int k = 0; k < KNNK; ++k) {
          const int j = knn[(abase + i) * KNNK + k];
          if (j >= 0) As[tid * SROW + j] = di * sdis[j];
        }
      }
      wait_async();                            // Xs panel resident (h==0)
      __syncthreads();

      // ---- GEMM: 8 i-tiles x CG c-groups, all operands from LDS
      for (int g = wave; g < 8 * CG; g += 8) {
        const int il0 = (g / CG) * 16;
        const int cg0 = (g % CG) * 64;
        v8f acc[4] = {};
#pragma unroll 2
        for (int j0 = 0; j0 < NN; j0 += 4) {
          const v2f a = *(const v2f*)&As[(il0 + ln) * SROW + j0 + 2 * lh];
#pragma unroll
          for (int cc = 0; cc < 4; ++cc) {
            const v2f bf =
                *(const v2f*)&Xs[(cg0 + cc * 16 + ln) * SROW + j0 + 2 * lh];
            acc[cc] = wmma_f32(a, bf, acc[cc]);
          }
        }
#pragma unroll
        for (int cc = 0; cc < 4; ++cc) {
          const int c = ch * CH + cg0 + cc * 16 + ln;
#pragma unroll
          for (int r = 0; r < 8; ++r) {
            const int i = h * 128 + il0 + r + 8 * lh;
            out[((b * C + c) * NN + i) * TC + t] = acc[cc][r];
          }
        }
      }
      __syncthreads();
    }
  }
}

// ---------------------------------------------------------------------------
extern "C" void kernel_launch(void* const* d_in, const int* in_sizes, int n_in,
                              void* d_out, int out_size, void* d_ws, size_t ws_size,
                              hipStream_t stream) {
  (void)in_sizes; (void)n_in; (void)out_size; (void)ws_size;
  const float* x  = (const float*)d_in[0];
  const float* W1 = (const float*)d_in[1];  const float* b1 = (const float*)d_in[2];
  const float* W2 = (const float*)d_in[3];  const float* b2 = (const float*)d_in[4];
  const float* W3 = (const float*)d_in[5];  const float* b3 = (const float*)d_in[6];
  const float* W4 = (const float*)d_in[7];  const float* b4 = (const float*)d_in[8];
  const float* W5 = (const float*)d_in[9];  const float* b5 = (const float*)d_in[10];

  char* ws = (char*)d_ws;
  const size_t ACT_BYTES = (size_t)32 * 64 * 256 * 16 * 4;   // 33.5 MB max act
  float* bufA = (float*)(ws);
  float* bufB = (float*)(ws + ACT_BYTES);
  int*   knn  = (int*)  (ws + 2 * ACT_BYTES);
  float* dinv = (float*)(ws + 2 * ACT_BYTES + (size_t)BB * TT * NN * KNNK * 4);
  float* dis  = dinv + (size_t)BB * TT * NN;

  const dim3 blk(256);
  adj_kernel<<<BB * TT, blk, 0, stream>>>(x, knn, dinv, dis);

  conv_kernel<4,   64, 1, 16, 16><<<BB * 16, blk, 0, stream>>>(x,    W1, b1, bufA);
  graph_kernel<64, 16>            <<<BB * 16, blk, 0, stream>>>(bufA, knn, dinv, dis, bufB);
  conv_kernel<64,  64, 1, 16, 16><<<BB * 16, blk, 0, stream>>>(bufB, W2, b2, bufA);
  graph_kernel<64, 16>            <<<BB * 16, blk, 0, stream>>>(bufA, knn, dinv, dis, bufB);
  conv_kernel<64, 128, 2, 16,  8><<<BB *  8, blk, 0, stream>>>(bufB, W3, b3, bufA);
  graph_kernel<128, 8>            <<<BB *  8, blk, 0, stream>>>(bufA, knn, dinv, dis, bufB);
  conv_kernel<128,128, 1,  8,  8><<<BB *  8, blk, 0, stream>>>(bufB, W4, b4, bufA);
  graph_kernel<128, 8>            <<<BB *  8, blk, 0, stream>>>(bufA, knn, dinv, dis, bufB);
  conv_kernel<128,256, 2,  8,  4><<<BB *  4, blk, 0, stream>>>(bufB, W5, b5, bufA);
  graph_kernel<256, 4>            <<<BB *  4, blk, 0, stream>>>(bufA, knn, dinv, dis, (float*)d_out);
}